// AtomWiseInvariants_38809324487061
// MI455X (gfx1250) — compile-verified
//
#include <hip/hip_runtime.h>

// CDNA5 / gfx1250: wave32, WMMA f32 16x16x4 (the only fp32 matrix op).
typedef float v2f __attribute__((ext_vector_type(2)));
typedef float v8f __attribute__((ext_vector_type(8)));

#define C_DIM   128
#define RBF_DIM 20

__device__ __forceinline__ v8f wmma_f32_k4(v2f a, v2f b, v8f c) {
  // 8 args: (neg_a, A, neg_b, B, c_mod, C, reuse_a, reuse_b)
  return __builtin_amdgcn_wmma_f32_16x16x4_f32(false, a, false, b, (short)0, c,
                                               false, false);
}

__device__ __forceinline__ float silu(float x) {
  return x / (1.0f + __expf(-x));
}

// ---------------------------------------------------------------------------
// Kernel 1: zero the [N, C] scatter accumulator (must run every launch).
// Regular-temporal stores: we WANT accum resident in the 192MB L2 for the
// atomic storm that follows.
// ---------------------------------------------------------------------------
__global__ __launch_bounds__(256) void zero_kernel(float4* __restrict__ p,
                                                   long long n4) {
  long long i = (long long)blockIdx.x * 256 + threadIdx.x;
  if (i < n4) p[i] = make_float4(0.f, 0.f, 0.f, 0.f);
}

// ---------------------------------------------------------------------------
// Kernel 2: per-edge  msg = ((rbf @ W_rbf.T + b) * env) * x_scalar
//           scatter-add into accum[edge_index_0[e], :].
// Block = 256 thr = 8 waves; block covers 128 edges, wave covers 16 edges.
// rbf GEMM tile (16x20 @ 20x128) via v_wmma_f32_16x16x4_f32, W_rbf in LDS.
//
// All global accesses use 32-bit unsigned element offsets from a uniform
// base (x_scalar spans 410MB < 2^31 bytes, rbf 64MB, accum 25.6MB), so the
// compiler can emit SADDR-form global_load/global_atomic with one 32-bit
// VGPR offset + constant immediate — no per-iteration 64-bit address math.
// Streaming inputs (rbf, x_scalar) are loaded non-temporally so the L2 keeps
// the accum working set instead.
//
// OOB robustness without branches: env is zeroed for rows >= E, so the
// contribution is exactly 0.0f and the (clamped-address) atomicAdd is a
// numeric no-op. All loads are clamped to row E-1.
// ---------------------------------------------------------------------------
__global__ __launch_bounds__(256) void edge_kernel(
    const float* __restrict__ x_scalar,  // [E, C]
    const float* __restrict__ rbf,       // [E, RBF]
    const float* __restrict__ env,       // [E, 1]
    const int*   __restrict__ eidx,      // [E]
    const float* __restrict__ W_rbf,     // [C, RBF]
    const float* __restrict__ b_rbf,     // [C]
    float* __restrict__ accum,           // [N, C]
    int E) {
  __shared__ float sW[C_DIM * RBF_DIM];  // 10 KB, W_rbf row-major [c][k]
  __shared__ float sBias[C_DIM];
  __shared__ float sEnv[128];
  __shared__ int   sIdx[128];

  const int tid = threadIdx.x;
  const int blockEdge0 = blockIdx.x * 128;

  for (int i = tid; i < C_DIM * RBF_DIM; i += 256) sW[i] = W_rbf[i];
  if (tid < C_DIM) sBias[tid] = b_rbf[tid];
  if (tid < 128) {
    int e  = blockEdge0 + tid;
    int ec = (e < E) ? e : (E - 1);
    sEnv[tid] = (e < E) ? env[ec] : 0.0f;  // OOB rows contribute exactly 0
    sIdx[tid] = eidx[ec];                  // clamped -> valid address
  }
  __syncthreads();

  const int wave  = tid >> 5;
  const int lane  = tid & 31;
  const int row   = lane & 15;   // M (A) / N (B,D) index
  const int khalf = lane >> 4;   // 0 -> K{0,1}, 1 -> K{2,3}

  const int eBase = blockEdge0 + wave * 16;
  const int eRow  = eBase + row;
  const int eRowC = (eRow < E) ? eRow : (E - 1);

  // A fragments: 5 K-steps of the 16x4 fp32 A layout (clamped rows are
  // annihilated by env == 0 in the epilogue). Non-temporal: rbf is
  // streamed exactly once.
  v2f a[5];
  const unsigned rbase = (unsigned)eRowC * RBF_DIM + 2 * khalf;
#pragma unroll
  for (int kk = 0; kk < 5; ++kk) {
    a[kk].x = __builtin_nontemporal_load(rbf + rbase + kk * 4);
    a[kk].y = __builtin_nontemporal_load(rbf + rbase + kk * 4 + 1);
  }

  // Per-row epilogue state as 32-bit element offsets (pre-offset by this
  // lane's column `row`): later accesses are base + const immediate.
  float    envv[8];
  unsigned xoff[8];
  unsigned doff[8];
#pragma unroll
  for (int g = 0; g < 8; ++g) {
    const int m  = g + 8 * khalf;
    const int e  = eBase + m;
    const int ec = (e < E) ? e : (E - 1);
    envv[g] = sEnv[wave * 16 + m];
    xoff[g] = (unsigned)ec * C_DIM + row;
    doff[g] = (unsigned)sIdx[wave * 16 + m] * C_DIM + row;
    // Pull this edge's 512B x_scalar row into the WGP caches while the
    // WMMAs run: 16 lanes x 8 floats covers the whole row. Locality 3 ->
    // nearest scope.
    __builtin_prefetch(x_scalar + (unsigned)ec * C_DIM + row * 8, 0, 3);
  }

  const float* sWrow = &sW[row * RBF_DIM + 2 * khalf];
  const float* sBrow = &sBias[row];

#pragma unroll
  for (int nb = 0; nb < 8; ++nb) {  // 8 x 16 = 128 output channels
    v8f d = {};
#pragma unroll
    for (int kk = 0; kk < 5; ++kk) {
      // offset nb*16*RBF_DIM and kk*4 are compile-time immediates
      v2f b;
      b.x = sWrow[nb * 16 * RBF_DIM + kk * 4];
      b.y = sWrow[nb * 16 * RBF_DIM + kk * 4 + 1];
      d = wmma_f32_k4(a[kk], b, d);
    }
    const float bias = sBrow[nb * 16];

    // Phase 1: issue all 8 x_scalar loads (saddr + immediate, non-temporal).
    float xv[8];
#pragma unroll
    for (int g = 0; g < 8; ++g)
      xv[g] = __builtin_nontemporal_load(x_scalar + xoff[g] + nb * 16);

    // Phase 2: modulate and scatter (hardware global_atomic_add_f32).
#pragma unroll
    for (int g = 0; g < 8; ++g) {
      float v = (d[g] + bias) * envv[g] * xv[g];
      unsafeAtomicAdd(accum + doff[g] + nb * 16, v);
    }
  }
}

// ---------------------------------------------------------------------------
// Kernel 3: per-atom MLP head.
//   h1 = silu(accum @ W1.T + b1); h2 = silu(h1 @ W2.T + b2); out = h2 @ W3.T + b3
// Block = 256 thr = 8 waves; 64 atoms per block. Each wave owns one 16-col
// slab of the 128 outputs; K=128 -> 32 WMMAs per 16x16 tile.
// ---------------------------------------------------------------------------
__global__ __launch_bounds__(256) void mlp_kernel(
    const float* __restrict__ accum,  // [N, C]
    const float* __restrict__ W1, const float* __restrict__ b1,
    const float* __restrict__ W2, const float* __restrict__ b2,
    const float* __restrict__ W3, const float* __restrict__ b3,
    float* __restrict__ out, int N) {
  __shared__ float sA[64 * C_DIM];  // 32 KB: layer input (reused for h2)
  __shared__ float sH[64 * C_DIM];  // 32 KB: layer-1 output
  __shared__ float sB1[C_DIM], sB2[C_DIM], sW3[C_DIM];

  const int tid   = threadIdx.x;
  const int atom0 = blockIdx.x * 64;

  if (tid < C_DIM) { sB1[tid] = b1[tid]; sB2[tid] = b2[tid]; sW3[tid] = W3[tid]; }
  for (int i = tid; i < 64 * C_DIM; i += 256) {
    const int atom = atom0 + (i >> 7);
    const int ac   = (atom < N) ? atom : 0;
    float v = accum[(unsigned)ac * C_DIM + (i & 127)];
    sA[i] = (atom < N) ? v : 0.0f;
  }

  const int wave  = tid >> 5;
  const int lane  = tid & 31;
  const int row   = lane & 15;
  const int khalf = lane >> 4;
  const int ncol  = wave * 16 + row;  // this lane's output channel

  // Warm this wave's W1/W2 row slabs (512B rows, spread across lanes).
  const unsigned wrow = (unsigned)ncol * C_DIM + khalf * 64;
  __builtin_prefetch(W1 + wrow, 0, 3);
  __builtin_prefetch(W1 + wrow + 32, 0, 3);
  __builtin_prefetch(W2 + wrow, 0, 3);
  __builtin_prefetch(W2 + wrow + 32, 0, 3);

  __syncthreads();

  // ---- layer 1: sA @ W1.T -> silu -> sH
  for (int t = 0; t < 4; ++t) {
    v8f d = {};
#pragma unroll 8
    for (int kk = 0; kk < 32; ++kk) {
      const float* ap = &sA[(t * 16 + row) * C_DIM + kk * 4 + 2 * khalf];
      v2f a; a.x = ap[0]; a.y = ap[1];
      const unsigned bo = (unsigned)ncol * C_DIM + kk * 4 + 2 * khalf;
      v2f b; b.x = W1[bo]; b.y = W1[bo + 1];
      d = wmma_f32_k4(a, b, d);
    }
#pragma unroll
    for (int g = 0; g < 8; ++g) {
      const int m = g + 8 * khalf;
      sH[(t * 16 + m) * C_DIM + ncol] = silu(d[g] + sB1[ncol]);
    }
  }
  __syncthreads();

  // ---- layer 2: sH @ W2.T -> silu -> sA (reuse)
  for (int t = 0; t < 4; ++t) {
    v8f d = {};
#pragma unroll 8
    for (int kk = 0; kk < 32; ++kk) {
      const float* ap = &sH[(t * 16 + row) * C_DIM + kk * 4 + 2 * khalf];
      v2f a; a.x = ap[0]; a.y = ap[1];
      const unsigned bo = (unsigned)ncol * C_DIM + kk * 4 + 2 * khalf;
      v2f b; b.x = W2[bo]; b.y = W2[bo + 1];
      d = wmma_f32_k4(a, b, d);
    }
#pragma unroll
    for (int g = 0; g < 8; ++g) {
      const int m = g + 8 * khalf;
      sA[(t * 16 + m) * C_DIM + ncol] = silu(d[g] + sB2[ncol]);
    }
  }
  __syncthreads();

  // ---- final 128 -> 1 projection (T == 1)
  if (tid < 64) {
    const int atom = atom0 + tid;
    if (atom < N) {
      float s = b3[0];
#pragma unroll 8
      for (int c = 0; c < C_DIM; ++c) s += sA[tid * C_DIM + c] * sW3[c];
      out[atom] = s;
    }
  }
}

// ---------------------------------------------------------------------------
extern "C" void kernel_launch(void* const* d_in, const int* in_sizes, int n_in,
                              void* d_out, int out_size, void* d_ws,
                              size_t ws_size, hipStream_t stream) {
  const float* x_scalar = (const float*)d_in[0];
  const float* rbf      = (const float*)d_in[1];
  const float* env      = (const float*)d_in[2];
  const int*   eidx     = (const int*)d_in[3];
  // d_in[4] = num_atoms scalar on device; N == out_size since T == 1.
  const float* W_rbf = (const float*)d_in[5];
  const float* b_rbf = (const float*)d_in[6];
  const float* W1 = (const float*)d_in[7];
  const float* b1 = (const float*)d_in[8];
  const float* W2 = (const float*)d_in[9];
  const float* b2 = (const float*)d_in[10];
  const float* W3 = (const float*)d_in[11];
  const float* b3 = (const float*)d_in[12];

  float* out   = (float*)d_out;
  float* accum = (float*)d_ws;  // [N, C] fp32 scatter accumulator (25.6 MB)

  const int E = in_sizes[3];
  const int N = out_size;  // T == 1

  const long long n4 = (long long)N * C_DIM / 4;  // C_DIM % 4 == 0
  zero_kernel<<<(unsigned)((n4 + 255) / 256), 256, 0, stream>>>(
      (float4*)accum, n4);

  edge_kernel<<<(E + 127) / 128, 256, 0, stream>>>(
      x_scalar, rbf, env, eidx, W_rbf, b_rbf, accum, E);

  mlp_kernel<<<(N + 63) / 64, 256, 0, stream>>>(
      accum, W1, b1, W2, b2, W3, b3, out, N);
}